// VolumeRenderer_74887049773540
// MI455X (gfx1250) — compile-verified
//
#include <hip/hip_runtime.h>
#include <hip/hip_bf16.h>
#include <stdint.h>

// Volume rendering (NeRF-style) ray integration.
// N = 65536 rays, L = 192 samples/ray.
// Memory-bound: ~251.5 MB read @ 23.3 TB/s => ~10.8 us floor. No WMMA
// formulation exists (per-ray weighted reductions share no operand matrix);
// the CDNA5-specific path exercised here is ASYNC global->LDS DMA (ASYNCcnt)
// for the dominant rgb stream + non-temporal hints + global_prefetch_b8.

#define RAY_L       192
#define CHUNKS      6                 // 192 / 32 lanes
#define RGB_FLOATS  (RAY_L * 3)       // 576 floats = 2304 B per ray
#define B64_ITERS   9                 // 2304 B / 8 B / 32 lanes
#define WAVES_PER_BLOCK 8

#if defined(__HIP_DEVICE_COMPILE__) && __has_builtin(__builtin_amdgcn_global_load_async_to_lds_b64)
#define USE_ASYNC_LDS 1
#else
#define USE_ASYNC_LDS 0
#endif

#if defined(__HIP_DEVICE_COMPILE__)
// b64 async DMA moves an 8-byte (v2i32) element: global AS(1) -> LDS AS(3).
typedef int v2i __attribute__((ext_vector_type(2)));
typedef __attribute__((address_space(1))) v2i* gv2i_t;
typedef __attribute__((address_space(3))) v2i* lv2i_t;
#endif

__device__ __forceinline__ float fast_sigmoid(float x) {
    // 1 / (1 + e^-x) via v_exp_f32 + v_rcp_f32
    return __builtin_amdgcn_rcpf(1.0f + __expf(-x));
}

__global__ __launch_bounds__(256)
void volrender_kernel(const float* __restrict__ depth,   // [N, L]
                      const float* __restrict__ rgb,     // [N, L, 3]
                      const float* __restrict__ sigma,   // [N, L]
                      float* __restrict__ color_out,     // [N, 3]
                      float* __restrict__ depth_out,     // [N]
                      int nrays)
{
    __shared__ float lds_rgb[WAVES_PER_BLOCK * RGB_FLOATS];   // 18432 B / block

    const int lane = threadIdx.x & 31;
    const int wib  = threadIdx.x >> 5;                        // wave id in block
    const int ray  = blockIdx.x * WAVES_PER_BLOCK + wib;
    if (ray >= nrays) return;                                 // uniform per wave

    const float* dptr = depth + (size_t)ray * RAY_L;
    const float* sptr = sigma + (size_t)ray * RAY_L;
    const float* gptr = rgb   + (size_t)ray * RGB_FLOATS;
    float*       lrgb = &lds_rgb[wib * RGB_FLOATS];

#if USE_ASYNC_LDS
    // Stage this ray's rgb (2304 B) into LDS with async DMA: 9 x b64 per lane,
    // lanes cover consecutive 8B chunks -> fully coalesced 256B/instr.
    #pragma unroll
    for (int i = 0; i < B64_ITERS; ++i) {
        const int byte_off = (i * 32 + lane) * 8;
        __builtin_amdgcn_global_load_async_to_lds_b64(
            (gv2i_t)((const char*)gptr + byte_off),
            (lv2i_t)((char*)lrgb + byte_off),
            0, 0);
    }
#endif

    // Prefetch depth/sigma streams (emits global_prefetch_b8; hint only).
    __builtin_prefetch(dptr + lane * 6, 0, 0);
    __builtin_prefetch(sptr + lane * 6, 0, 0);

    // ---- Phase A: weights + depth accumulation (overlaps the async DMA) ----
    float wv[CHUNKS];
    float accR = 0.0f, accG = 0.0f, accB = 0.0f, accD = 0.0f;
    #pragma unroll
    for (int i = 0; i < CHUNKS; ++i) {
        const int   l     = i * 32 + lane;
        const float dl    = __builtin_nontemporal_load(dptr + l);
        const float sg    = __builtin_nontemporal_load(sptr + l);
        const int   ln    = (l < RAY_L - 1) ? (l + 1) : l;     // last bin pads
        const float dn    = dptr[ln];                          // L2/L0 re-hit
        const float delta = (l < RAY_L - 1) ? (dn - dl) : 1e10f;
        const float s     = fmaxf(sg, 0.0f);                   // relu
        const float alpha = 1.0f - __expf(-s * delta);
        const float w     = alpha * (1.0f - alpha + 1e-10f);   // cumprod(size1)=id
        wv[i] = w;
        accD += w * dl;
    }

#if USE_ASYNC_LDS
  #if __has_builtin(__builtin_amdgcn_s_wait_asynccnt)
    __builtin_amdgcn_s_wait_asynccnt(0);
  #else
    asm volatile("s_wait_asynccnt 0" ::: "memory");
  #endif
#endif

    // ---- Phase B: color accumulation from staged rgb ----
    #pragma unroll
    for (int i = 0; i < CHUNKS; ++i) {
        const int l = i * 32 + lane;
#if USE_ASYNC_LDS
        const float r = lrgb[l * 3 + 0];   // stride-3 dwords: conflict-free on 64 banks
        const float g = lrgb[l * 3 + 1];
        const float b = lrgb[l * 3 + 2];
#else
        const float r = __builtin_nontemporal_load(gptr + l * 3 + 0);
        const float g = __builtin_nontemporal_load(gptr + l * 3 + 1);
        const float b = __builtin_nontemporal_load(gptr + l * 3 + 2);
#endif
        const float w = wv[i];
        accR += w * fast_sigmoid(r);
        accG += w * fast_sigmoid(g);
        accB += w * fast_sigmoid(b);
    }

    // ---- wave32 tree reduction ----
    #pragma unroll
    for (int off = 16; off > 0; off >>= 1) {
        accR += __shfl_down(accR, off, 32);
        accG += __shfl_down(accG, off, 32);
        accB += __shfl_down(accB, off, 32);
        accD += __shfl_down(accD, off, 32);
    }

    if (lane == 0) {
        color_out[(size_t)ray * 3 + 0] = accR;
        color_out[(size_t)ray * 3 + 1] = accG;
        color_out[(size_t)ray * 3 + 2] = accB;
        depth_out[ray]                 = accD;
    }
}

extern "C" void kernel_launch(void* const* d_in, const int* in_sizes, int n_in,
                              void* d_out, int out_size, void* d_ws, size_t ws_size,
                              hipStream_t stream) {
    const float* depth = (const float*)d_in[0];   // [N, L, 1]
    const float* rgb   = (const float*)d_in[1];   // [N, L, 3]
    const float* sigma = (const float*)d_in[2];   // [N, L, 1]

    const int nrays = in_sizes[0] / RAY_L;        // N = 65536

    float* out       = (float*)d_out;
    float* color_out = out;                        // [N,3] flat, first output
    float* depth_out = out + (size_t)nrays * 3;    // [N,1] second output

    const int grid = (nrays + WAVES_PER_BLOCK - 1) / WAVES_PER_BLOCK;  // 8192
    hipLaunchKernelGGL(volrender_kernel, dim3(grid), dim3(256), 0, stream,
                       depth, rgb, sigma, color_out, depth_out, nrays);
}